// Project_28132035789035
// MI455X (gfx1250) — compile-verified
//
#include <hip/hip_runtime.h>
#include <stdint.h>

// Reference constants (HEIGHT, WIDTH, B, N from the reference module).
#define KH 480
#define KW 640
#define KB 4
#define KN 500000
#define HWSZ (KH * KW)

#define TILE 16
#define PAD 2
#define TDH (TILE + 2 * PAD)    // 20 tile rows for k=5
#define TDW 24                  // padded tile width: 6 aligned b128 segments/row
#define FT  (TILE + 2)          // 18x18 halo tile for 3x3 conv

// ---------------------------------------------------------------------------
// CDNA5 async global -> LDS copies (GVS form: SGPR base + per-lane i32 offset).
// LDS address comes from the VDST VGPR; tracked with ASYNCcnt (ISA 10.7 / 15.18.3).
// ---------------------------------------------------------------------------
__device__ __forceinline__ void async_b32_to_lds(unsigned ldsByteOff,
                                                 const float* __restrict__ base,
                                                 unsigned gByteOff) {
  asm volatile("global_load_async_to_lds_b32 %0, %1, %2"
               :: "v"(ldsByteOff), "v"(gByteOff), "s"(base)
               : "memory");
}

__device__ __forceinline__ void async_b128_to_lds(unsigned ldsByteOff,
                                                  const float* __restrict__ base,
                                                  unsigned gByteOff) {
  // Both addresses must be 16B aligned (guaranteed by caller).
  asm volatile("global_load_async_to_lds_b128 %0, %1, %2"
               :: "v"(ldsByteOff), "v"(gByteOff), "s"(base)
               : "memory");
}

__device__ __forceinline__ void wait_async_lds() {
  asm volatile("s_wait_asynccnt 0" ::: "memory");
}

__device__ __forceinline__ int reflect_idx(int i, int n) {
  // jnp.pad mode='reflect' (no edge repeat); pad <= 4 so one fold suffices.
  if (i < 0) i = -i;
  if (i >= n) i = 2 * n - 2 - i;
  return i;
}

// ---------------------------------------------------------------------------
// Kernel 0: init depth buffer + image background
// ---------------------------------------------------------------------------
__global__ void k_init(int* __restrict__ depth, float* __restrict__ img) {
  int i = blockIdx.x * blockDim.x + threadIdx.x;
  if (i < KB * 3 * HWSZ) img[i] = -0.001f;
  if (i < KB * HWSZ)     depth[i] = 0x7fffffff;
}

// ---------------------------------------------------------------------------
// Kernel 1: per-batch Tinv (adjugate), then P = K * Tinv[0:3,:], D = Tinv[2,:]
// ---------------------------------------------------------------------------
__device__ float det3x3(const float* m, int r0, int r1, int r2,
                        int c0, int c1, int c2) {
  return m[r0 * 4 + c0] * (m[r1 * 4 + c1] * m[r2 * 4 + c2] - m[r1 * 4 + c2] * m[r2 * 4 + c1])
       - m[r0 * 4 + c1] * (m[r1 * 4 + c0] * m[r2 * 4 + c2] - m[r1 * 4 + c2] * m[r2 * 4 + c0])
       + m[r0 * 4 + c2] * (m[r1 * 4 + c0] * m[r2 * 4 + c1] - m[r1 * 4 + c1] * m[r2 * 4 + c0]);
}

__global__ void k_prep(const float* __restrict__ Kc, const float* __restrict__ Tc,
                       float* __restrict__ PD) {
  int b = blockIdx.x * blockDim.x + threadIdx.x;
  if (b >= KB) return;
  float m[16], inv[16], cof[16];
  for (int i = 0; i < 16; ++i) m[i] = Tc[b * 16 + i];
  const int idxs[4][3] = {{1,2,3},{0,2,3},{0,1,3},{0,1,2}};
  for (int i = 0; i < 4; ++i)
    for (int j = 0; j < 4; ++j) {
      float d = det3x3(m, idxs[i][0], idxs[i][1], idxs[i][2],
                          idxs[j][0], idxs[j][1], idxs[j][2]);
      cof[i * 4 + j] = ((i + j) & 1) ? -d : d;
    }
  float det = m[0] * cof[0] + m[1] * cof[1] + m[2] * cof[2] + m[3] * cof[3];
  float r = 1.0f / det;
  for (int i = 0; i < 4; ++i)
    for (int j = 0; j < 4; ++j)
      inv[i * 4 + j] = cof[j * 4 + i] * r;   // adjugate transpose
  const float* kb = Kc + b * 9;
  float* p = PD + b * 16;
  for (int rI = 0; rI < 3; ++rI)
    for (int c = 0; c < 4; ++c)
      p[rI * 4 + c] = kb[rI * 3 + 0] * inv[0 * 4 + c]
                    + kb[rI * 3 + 1] * inv[1 * 4 + c]
                    + kb[rI * 3 + 2] * inv[2 * 4 + c];
  for (int c = 0; c < 4; ++c) p[12 + c] = inv[2 * 4 + c];
}

// ---------------------------------------------------------------------------
// Per-point projection helper
// ---------------------------------------------------------------------------
__device__ __forceinline__ void project_point(const float* __restrict__ cloud,
                                              const float* __restrict__ PD,
                                              long long i, int& b, int& lin, int& fd) {
  b = (int)(i / KN);
  int n = (int)(i % KN);
  const float* c = cloud + (size_t)b * 4 * KN + n;
  float X = c[0], Y = c[KN], Z = c[2 * KN], Wc = c[3 * KN];
  const float* p = PD + b * 16;
  float px = p[0] * X + p[1] * Y + p[2]  * Z + p[3]  * Wc;
  float py = p[4] * X + p[5] * Y + p[6]  * Z + p[7]  * Wc;
  float pz = p[8] * X + p[9] * Y + p[10] * Z + p[11] * Wc;
  float d  = p[12] * X + p[13] * Y + p[14] * Z + p[15] * Wc;
  int u = (int)(px / pz); u = min(max(u, 0), KW - 1);
  int v = (int)(py / pz); v = min(max(v, 0), KH - 1);
  fd  = (int)d;                       // astype(int32): truncate toward zero
  lin = b * HWSZ + v * KW + u;
}

// Kernel 2: z-buffer atomic min
__global__ void k_zmin(const float* __restrict__ cloud, const float* __restrict__ PD,
                       int* __restrict__ depth) {
  long long i = (long long)blockIdx.x * blockDim.x + threadIdx.x;
  if (i >= (long long)KB * KN) return;
  int b, lin, fd;
  project_point(cloud, PD, i, b, lin, fd);
  atomicMin(&depth[lin], fd);
}

// Kernel 3: winners scatter RGB
__global__ void k_zset(const float* __restrict__ cloud, const float* __restrict__ rgbv,
                       const float* __restrict__ PD, const int* __restrict__ depth,
                       float* __restrict__ img) {
  long long i = (long long)blockIdx.x * blockDim.x + threadIdx.x;
  if (i >= (long long)KB * KN) return;
  int b, lin, fd;
  project_point(cloud, PD, i, b, lin, fd);
  if (fd == depth[lin]) {
    int n = (int)(i % KN);
    int pix = lin - b * HWSZ;
    const float* rgb = rgbv + (size_t)b * 3 * KN + n;
    float* ip = img + (size_t)b * 3 * HWSZ + pix;
    ip[0]        = rgb[0];
    ip[HWSZ]     = rgb[KN];
    ip[2 * HWSZ] = rgb[2 * KN];
  }
}

// Kernel 4: mask = (channel0 > 0), fixed for all 10 iterations
__global__ void k_mask(const float* __restrict__ img, unsigned char* __restrict__ mask) {
  int i = blockIdx.x * blockDim.x + threadIdx.x;
  if (i >= KB * HWSZ) return;
  int b = i / HWSZ, p = i % HWSZ;
  mask[i] = (img[(size_t)b * 3 * HWSZ + p] > 0.0f) ? 1 : 0;
}

// ---------------------------------------------------------------------------
// Kernel 5: masked median filter. k==5 fast path: async global->LDS halo tile
// (b128 segments for x-interior tiles, reflected b32 gather at x edges),
// then an unrolled partial selection to the 13th element (v_min/v_max only).
// ---------------------------------------------------------------------------
__global__ void __launch_bounds__(256)
k_median(const float* __restrict__ src, float* __restrict__ dst,
         const unsigned char* __restrict__ mask, const int* __restrict__ kptr) {
  const int k  = *kptr;                          // uniform across grid
  const int bx = blockIdx.x * TILE;
  const int by = blockIdx.y * TILE;
  const int bc = blockIdx.z;                     // b*3 + c
  const int b  = bc / 3;
  const int tx = threadIdx.x, ty = threadIdx.y;
  const int t  = ty * TILE + tx;
  const size_t plane = (size_t)bc * HWSZ;
  const int x = bx + tx, y = by + ty;

  if (k == 5) {
    // Tile layout: 20 rows x 24 cols; column lx maps to gx = bx - 4 + lx.
    // Pixel x=bx+tx sits at lx = tx+4; its 5-wide window spans lx = tx+2..tx+6.
    __shared__ __align__(16) float tile[TDH * TDW];
    unsigned ldsBase = (unsigned)(uintptr_t)(void*)tile;

    const bool interior = (bx >= 4) && (bx + TILE + 4 <= KW);
    if (interior) {
      // 20 rows x 6 b128 segments covering gx in [bx-4, bx+20).
      // Global byte addr (plane + gy*KW + bx-4+4s)*4 and LDS addr r*96+s*16
      // are both 16B aligned (bx % 16 == 0, plane/row byte strides % 16 == 0).
      if (t < TDH * 6) {
        int r = t / 6, s = t % 6;
        int gy = reflect_idx(by + r - PAD, KH);
        unsigned gOff = (unsigned)((plane + (size_t)gy * KW + (size_t)(bx - 4 + s * 4)) * 4u);
        unsigned lOff = ldsBase + (unsigned)(r * TDW + s * 4) * 4u;
        async_b128_to_lds(lOff, src, gOff);
      }
    } else {
      // x-edge tiles: per-element reflected gather into columns 2..21.
      for (int idx = t; idx < TDH * TDH; idx += TILE * TILE) {
        int r = idx / TDH, cx = idx % TDH;       // cx in 0..19 -> gx = bx-2+cx
        int gy = reflect_idx(by + r - PAD, KH);
        int gx = reflect_idx(bx + cx - PAD, KW);
        unsigned gOff = (unsigned)((plane + (size_t)gy * KW + gx) * 4u);
        unsigned lOff = ldsBase + (unsigned)(r * TDW + cx + 2) * 4u;
        async_b32_to_lds(lOff, src, gOff);
      }
    }
    wait_async_lds();      // this wave's async transfers complete (ASYNCcnt==0)
    __syncthreads();       // all waves' transfers visible in LDS

    float m[25];
    #pragma unroll
    for (int dy = 0; dy < 5; ++dy)
      #pragma unroll
      for (int dx = 0; dx < 5; ++dx)
        m[dy * 5 + dx] = tile[(ty + dy) * TDW + (tx + 2 + dx)];

    // Partial selection sort: after pass i, m[i] is the i-th smallest;
    // m[12] is the median of 25. 234 compare-exchanges, all fmin/fmax.
    #pragma unroll
    for (int i = 0; i <= 12; ++i) {
      #pragma unroll
      for (int j = i + 1; j < 25; ++j) {
        float a = m[i], bb = m[j];
        m[i] = fminf(a, bb);
        m[j] = fmaxf(a, bb);
      }
    }
    float cur = tile[(ty + PAD) * TDW + (tx + 4)];
    size_t o = plane + (size_t)y * KW + x;
    dst[o] = mask[(size_t)b * HWSZ + (size_t)y * KW + x] ? cur : m[12];
  } else {
    // Generic odd-k fallback (never hit for the reference's k=5).
    int pad = k >> 1;
    if (pad > 4) pad = 4;
    int kk = 2 * pad + 1;
    float w[81];
    int n = 0;
    for (int dy = -pad; dy <= pad; ++dy) {
      int gy = reflect_idx(y + dy, KH);
      for (int dx = -pad; dx <= pad; ++dx) {
        int gx = reflect_idx(x + dx, KW);
        w[n++] = src[plane + (size_t)gy * KW + gx];
      }
    }
    int mid = (kk * kk) / 2;
    for (int i = 0; i <= mid; ++i)
      for (int j = i + 1; j < n; ++j) {
        float a = w[i], c2 = w[j];
        w[i] = fminf(a, c2);
        w[j] = fmaxf(a, c2);
      }
    float cur = src[plane + (size_t)y * KW + x];
    size_t o = plane + (size_t)y * KW + x;
    dst[o] = mask[(size_t)b * HWSZ + (size_t)y * KW + x] ? cur : w[mid];
  }
}

// ---------------------------------------------------------------------------
// Kernel 6: single = any(c>0); 3x3 zero-padded sum; holemask = (sum>=6); mul
// ---------------------------------------------------------------------------
__global__ void __launch_bounds__(256)
k_final(const float* __restrict__ src, float* __restrict__ out) {
  __shared__ float sng[FT * FT];
  const int bx = blockIdx.x * TILE, by = blockIdx.y * TILE, b = blockIdx.z;
  const int tx = threadIdx.x, ty = threadIdx.y;
  const int t = ty * TILE + tx;
  const size_t base = (size_t)b * 3 * HWSZ;
  for (int idx = t; idx < FT * FT; idx += TILE * TILE) {
    int ly = idx / FT, lx = idx % FT;
    int gy = by + ly - 1, gx = bx + lx - 1;
    float s = 0.0f;
    if (gy >= 0 && gy < KH && gx >= 0 && gx < KW) {
      size_t o = base + (size_t)gy * KW + gx;
      s = (src[o] > 0.0f || src[o + HWSZ] > 0.0f || src[o + 2 * HWSZ] > 0.0f)
              ? 1.0f : 0.0f;
    }
    sng[idx] = s;
  }
  __syncthreads();
  const int x = bx + tx, y = by + ty;
  if (x < KW && y < KH) {
    float cnt = 0.0f;
    #pragma unroll
    for (int dy = 0; dy < 3; ++dy)
      #pragma unroll
      for (int dx = 0; dx < 3; ++dx)
        cnt += sng[(ty + dy) * FT + (tx + dx)];
    float hole = (cnt >= 6.0f) ? 1.0f : 0.0f;
    size_t o = base + (size_t)y * KW + x;
    out[o]            = src[o]            * hole;
    out[o + HWSZ]     = src[o + HWSZ]     * hole;
    out[o + 2 * HWSZ] = src[o + 2 * HWSZ] * hole;
  }
}

// ---------------------------------------------------------------------------
// Host orchestration (graph-capture safe: only kernel launches on `stream`)
// ---------------------------------------------------------------------------
extern "C" void kernel_launch(void* const* d_in, const int* in_sizes, int n_in,
                              void* d_out, int out_size, void* d_ws, size_t ws_size,
                              hipStream_t stream) {
  const float* cloud = (const float*)d_in[0];   // (B,4,N)
  const float* rgbv  = (const float*)d_in[1];   // (B,3,N)
  const float* Kc    = (const float*)d_in[2];   // (B,3,3)
  const float* Tc    = (const float*)d_in[3];   // (B,4,4)
  const int*   kptr  = (const int*)d_in[4];     // scalar median_kernel_size

  char* w = (char*)d_ws;
  float* PD = (float*)w;
  size_t off = 1024;
  int* depth = (int*)(w + off);                    off += (size_t)4 * KB * HWSZ;
  unsigned char* mask = (unsigned char*)(w + off); off += (size_t)KB * HWSZ;
  float* img0 = (float*)(w + off);                 off += (size_t)4 * KB * 3 * HWSZ;
  float* img1 = (float*)(w + off);

  float* out = (float*)d_out;

  const int totalInit = KB * 3 * HWSZ;
  k_init<<<(totalInit + 255) / 256, 256, 0, stream>>>(depth, img0);
  k_prep<<<1, 32, 0, stream>>>(Kc, Tc, PD);

  const long long pts = (long long)KB * KN;
  const int pblocks = (int)((pts + 255) / 256);
  k_zmin<<<pblocks, 256, 0, stream>>>(cloud, PD, depth);
  k_zset<<<pblocks, 256, 0, stream>>>(cloud, rgbv, PD, depth, img0);
  k_mask<<<(KB * HWSZ + 255) / 256, 256, 0, stream>>>(img0, mask);

  dim3 mb(TILE, TILE, 1);
  dim3 mg(KW / TILE, KH / TILE, KB * 3);
  float* cur = img0;
  float* nxt = img1;
  for (int it = 0; it < 10; ++it) {
    k_median<<<mg, mb, 0, stream>>>(cur, nxt, mask, kptr);
    float* tswap = cur; cur = nxt; nxt = tswap;
  }
  dim3 fg(KW / TILE, KH / TILE, KB);
  k_final<<<fg, mb, 0, stream>>>(cur, out);
}